// LSTMRecurrentModel_27092653703631
// MI455X (gfx1250) — compile-verified
//
#include <hip/hip_runtime.h>
#include <cmath>
#include <cstdint>

// Problem constants (match reference)
#define L      16384
#define DD     1024
#define FOURD  4096

// Recurrent-phase configuration
#define NWG        64   // persistent workgroups; WG k owns h-dims [16k, 16k+16)
#define HDIMS      16   // h dims per WG
#define COLS       64   // 16 h-dims x 4 gates = Wh columns per WG
#define THREADS    256
#define WPAD       1025 // padded LDS column stride (floats) to spread banks

typedef __attribute__((ext_vector_type(2))) float v2f;
typedef __attribute__((ext_vector_type(8))) float v8f;

// ---------------------------------------------------------------------------
// Phase 1: zx = xs @ Wi   [16384 x 1024] * [1024 x 4096] -> [16384 x 4096]
// fp32 WMMA 16x16x4. Each wave computes a 16x64 tile (4 accumulators),
// reusing the A fragment across the 4 N-subtiles. Wi (16MB) is L2-resident,
// so direct global feeds are sufficient on MI455X (192MB L2, 23.3 TB/s HBM).
// ---------------------------------------------------------------------------
__global__ __launch_bounds__(256) void gemm_zx_kernel(
    const float* __restrict__ xs,   // [L][DD]
    const float* __restrict__ Wi,   // [DD][FOURD]
    float* __restrict__ zx)         // [L][FOURD]
{
  const int waveId = blockIdx.x * (blockDim.x >> 5) + (threadIdx.x >> 5);
  const int mTile  = waveId >> 6;        // 0..1023  (L/16)
  const int n64    = waveId & 63;        // 0..63    (FOURD/64)
  const int m0     = mTile * 16;
  const int n0     = n64 * 64;

  const int lane = threadIdx.x & 31;
  const int half = lane >> 4;            // 0: K 0..1 / rows 0..7 ; 1: K 2..3 / rows 8..15
  const int lm   = lane & 15;

  v8f acc0 = {}, acc1 = {}, acc2 = {}, acc3 = {};

  const float* arow = xs + (size_t)(m0 + lm) * DD + 2 * half;
  for (int k0 = 0; k0 < DD; k0 += 4) {
    // A fragment: 16x4 f32. lane<16: K=k0,k0+1 ; lane>=16: K=k0+2,k0+3
    v2f a = *(const v2f*)(arow + k0);
    const float* brow0 = Wi + (size_t)(k0 + 2 * half) * FOURD + n0 + lm;
    // B fragments: 4x16 f32 per N-subtile. b.x=B[2h][n], b.y=B[2h+1][n]
    {
      v2f b; b.x = brow0[0];      b.y = brow0[FOURD];
      acc0 = __builtin_amdgcn_wmma_f32_16x16x4_f32(false, a, false, b, (short)0, acc0, false, false);
    }
    {
      v2f b; b.x = brow0[16];     b.y = brow0[FOURD + 16];
      acc1 = __builtin_amdgcn_wmma_f32_16x16x4_f32(false, a, false, b, (short)0, acc1, false, false);
    }
    {
      v2f b; b.x = brow0[32];     b.y = brow0[FOURD + 32];
      acc2 = __builtin_amdgcn_wmma_f32_16x16x4_f32(false, a, false, b, (short)0, acc2, false, false);
    }
    {
      v2f b; b.x = brow0[48];     b.y = brow0[FOURD + 48];
      acc3 = __builtin_amdgcn_wmma_f32_16x16x4_f32(false, a, false, b, (short)0, acc3, false, false);
    }
  }

  // C/D layout: VGPR v -> row (v + half*8), col n0 + lm
  for (int v = 0; v < 8; ++v) {
    const size_t row = (size_t)(m0 + v + half * 8) * FOURD + n0 + lm;
    zx[row +  0] = acc0[v];
    zx[row + 16] = acc1[v];
    zx[row + 32] = acc2[v];
    zx[row + 48] = acc3[v];
  }
}

// ---------------------------------------------------------------------------
// Phase 2: persistent LSTM recurrence. 64 WGs; each caches its 64 Wh columns
// (1024 x 64 fp32 = 256KB) in LDS. Per step: async global->LDS broadcast of
// prev h, 64 length-1024 dot products (float4 LDS feeds), gate math (c-state
// in registers of lanes 0-15), h slice written into ys[t], device grid barrier.
// zx[t+1] is software-pipelined across the barrier to hide its L2 latency.
// ---------------------------------------------------------------------------
__device__ __forceinline__ float sigmoid_f(float x) {
  return 1.0f / (1.0f + __expf(-x));
}

__global__ __launch_bounds__(THREADS) void lstm_recurrent_kernel(
    const float* __restrict__ zx,   // [L][FOURD]
    const float* __restrict__ Wh,   // [DD][FOURD]
    const float* __restrict__ b,    // [FOURD]
    float* __restrict__ out,        // [L][DD] == h history
    unsigned* barrier_cnt,
    unsigned* barrier_gen)
{
  extern __shared__ float smem[];
  float* w       = smem;                       // [COLS][WPAD]
  float* hbuf    = smem + COLS * WPAD;         // [DD]
  float* partial = hbuf + DD;                  // [THREADS]
  float* zcol    = partial + THREADS;          // [COLS]

  const int wg  = blockIdx.x;      // 0..63
  const int tid = threadIdx.x;     // 0..255

  const int c      = tid & (COLS - 1);  // column within WG
  const int rquart = tid >> 6;          // which quarter of K this thread sums
  const int gate   = c >> 4;            // 0..3 (i,f,g,o)
  const int hd     = c & (HDIMS - 1);
  const int absCol = gate * DD + wg * HDIMS + hd;

  // Stage this WG's Wh slice into LDS (one-time). 16-float contiguous global
  // segments; LDS writes land on distinct banks (WPAD is odd).
  for (int idx = tid; idx < COLS * DD; idx += THREADS) {
    int r  = idx >> 6;         // row 0..1023
    int cc = idx & (COLS - 1); // column 0..63
    int g2 = cc >> 4, h2 = cc & (HDIMS - 1);
    w[cc * WPAD + r] = Wh[(size_t)r * FOURD + g2 * DD + wg * HDIMS + h2];
  }
  const float bias = b[absCol];
  float cstate = 0.0f;          // lanes 0..15 own c-state for their h-dim

  // LDS byte offset of this lane's 16B chunk of hbuf (for async global->LDS)
  const unsigned hbuf_lds = (unsigned)(uintptr_t)(void*)(hbuf + tid * 4);

  __syncthreads();

  // Software-pipelined zx feed: zcur holds zx[t][absCol]
  float zcur = zx[absCol];

  for (int t = 0; t < L; ++t) {
    // previous h -> LDS
    if (t == 0) {
      for (int r = tid; r < DD; r += THREADS) hbuf[r] = 0.0f;
    } else {
      // CDNA5 async copy: 256 lanes x B128 = full 4KB h row, global -> LDS,
      // tracked by ASYNCcnt (no VGPR round-trip on the serial path).
      const float* gsrc = out + (size_t)(t - 1) * DD + tid * 4;
      asm volatile("global_load_async_to_lds_b128 %0, %1, off\n\t"
                   "s_wait_asynccnt 0"
                   :: "v"(hbuf_lds), "v"(gsrc) : "memory");
    }
    __syncthreads();

    // GEMV: column c, rows [rquart*256, rquart*256+256)
    const float* wc = &w[c * WPAD + rquart * 256];
    const float* hc = &hbuf[rquart * 256];
    float acc = 0.0f;
#pragma unroll 8
    for (int r = 0; r < 256; r += 4) {
      float4 wv = *(const float4*)(wc + r);
      float4 hv = *(const float4*)(hc + r);
      acc = fmaf(wv.x, hv.x, acc);
      acc = fmaf(wv.y, hv.y, acc);
      acc = fmaf(wv.z, hv.z, acc);
      acc = fmaf(wv.w, hv.w, acc);
    }
    partial[tid] = acc;
    __syncthreads();

    // Prefetch next step's zx element; wait deferred to next iteration.
    const size_t tn = (t + 1 < L) ? (size_t)(t + 1) : 0;
    const float znext = zx[tn * FOURD + absCol];

    if (tid < COLS) {
      float z = partial[tid] + partial[tid + 64] + partial[tid + 128] + partial[tid + 192];
      z += zcur + bias;
      zcol[tid] = z;
    }
    __syncthreads();

    if (tid < HDIMS) {
      float zi = zcol[tid];
      float zf = zcol[tid + 16];
      float zg = zcol[tid + 32];
      float zo = zcol[tid + 48];
      float ig = sigmoid_f(zi);
      float fg = sigmoid_f(zf);
      float gg = tanhf(zg);
      float og = sigmoid_f(zo);
      cstate = fg * cstate + ig * gg;
      float h = og * tanhf(cstate);
      out[(size_t)t * DD + wg * HDIMS + tid] = h;
      __threadfence();   // h slice device-visible before barrier arrival
    }
    __syncthreads();

    // Device-scope grid barrier (sense counter on generation number).
    if (tid == 0) {
      unsigned arrived = __hip_atomic_fetch_add(barrier_cnt, 1u,
                            __ATOMIC_ACQ_REL, __HIP_MEMORY_SCOPE_AGENT);
      if (arrived == NWG - 1) {
        __hip_atomic_store(barrier_cnt, 0u,
                           __ATOMIC_RELAXED, __HIP_MEMORY_SCOPE_AGENT);
        __hip_atomic_fetch_add(barrier_gen, 1u,
                               __ATOMIC_RELEASE, __HIP_MEMORY_SCOPE_AGENT);
      } else {
        while (__hip_atomic_load(barrier_gen, __ATOMIC_ACQUIRE,
                                 __HIP_MEMORY_SCOPE_AGENT) < (unsigned)(t + 1)) {
          __builtin_amdgcn_s_sleep(1);
        }
      }
    }
    __syncthreads();

    zcur = znext;
  }
}

// ---------------------------------------------------------------------------
extern "C" void kernel_launch(void* const* d_in, const int* in_sizes, int n_in,
                              void* d_out, int out_size, void* d_ws, size_t ws_size,
                              hipStream_t stream) {
  const float* xs = (const float*)d_in[0];   // [L][DD]
  const float* Wi = (const float*)d_in[1];   // [DD][FOURD]
  const float* Wh = (const float*)d_in[2];   // [DD][FOURD]
  const float* b  = (const float*)d_in[3];   // [FOURD]
  float* out = (float*)d_out;                // [L][DD]

  float* zx = (float*)d_ws;                  // L*FOURD floats = 256MB scratch
  unsigned* barrier = (unsigned*)((char*)d_ws + (size_t)L * FOURD * sizeof(float));

  hipMemsetAsync(barrier, 0, 2 * sizeof(unsigned), stream);

  // Phase 1: 1024 M-tiles x 64 N64-tiles = 65536 waves -> 8192 blocks of 8 waves
  gemm_zx_kernel<<<8192, 256, 0, stream>>>(xs, Wi, zx);

  // Phase 2: persistent recurrence, 64 WGPs, ~267KB dynamic LDS each
  size_t lds_bytes = (size_t)(COLS * WPAD + DD + THREADS + COLS) * sizeof(float);
  lstm_recurrent_kernel<<<NWG, THREADS, lds_bytes, stream>>>(zx, Wh, b, out,
                                                             barrier, barrier + 1);
}